// NODE_auto_64544768525257
// MI455X (gfx1250) — compile-verified
//
#include <hip/hip_runtime.h>
#include <stdint.h>

// ---------------------------------------------------------------------------
// Neural-ODE (scalar state) on MI455X / gfx1250, wave32.
//
// Factorization: dyn-input [a, x, Uvec] has Uvec constant across all 124
// sequential evaluations, so G = Uvec @ Wd1[:,2:].T + bd1 is hoisted out of
// the time loop (turns ~25.5 GFLOP of serial GEMM into one 0.2 GFLOP WMMA
// GEMM). GEMMs use V_WMMA_F32_16X16X4_F32 at full fp32 precision. The time
// loop is latency-bound VALU/transcendental work, one wave per sample, with
// the shared 98->512 weight columns staged into LDS once per block via
// GLOBAL_LOAD_ASYNC_TO_LDS_B32 (ASYNCcnt / s_wait_asynccnt).
// ---------------------------------------------------------------------------

typedef __attribute__((ext_vector_type(2))) float v2f;
typedef __attribute__((ext_vector_type(8))) float v8f;

#define B_N    2048
#define R_N    32
#define D_INIT 96     // R*NMOD
#define H_INIT 256
#define D_DYN  98     // R*NMOD + 2
#define H_DYN  512
#define STEPS  32

// --- CDNA5 async copy: copies one float from global gsrc to the LDS address lds_addr, tracked by ASYNCcnt ----------
__device__ __forceinline__ void async_f32_to_lds(uint32_t lds_addr,
                                                 const float* gsrc) {
  asm volatile("global_load_async_to_lds_b32 %0, %1, off"
               :: "v"(lds_addr), "v"(gsrc)
               : "memory");
}
__device__ __forceinline__ void wait_asynccnt0() {
  asm volatile("s_wait_asynccnt 0x0" ::: "memory");
}

// ---------------------------------------------------------------------------
// Kernel 1: gather Uvec[b][0:96] = {U0[i0], U1[i1], U2[i2]}
// grid = B, block = 96 (3 waves)
// ---------------------------------------------------------------------------
__global__ void node_gather(const int* __restrict__ b_i_n,
                            const float* __restrict__ U0,
                            const float* __restrict__ U1,
                            const float* __restrict__ U2,
                            float* __restrict__ Uvec) {
  const int b   = blockIdx.x;
  const int j   = threadIdx.x;       // 0..95
  const int mod = j >> 5;            // 0..2
  const int r   = j & 31;
  const int idx = b_i_n[b * 3 + mod];
  const float* U = (mod == 0) ? U0 : (mod == 1) ? U1 : U2;
  Uvec[b * D_INIT + j] = U[idx * R_N + r];
}

// ---------------------------------------------------------------------------
// Kernel 2: x0[b] = tanh(Uvec @ Wi1.T + bi1) @ Wi2.T + bi2
// grid = B/16 = 128, block = 512 (16 waves); wave w owns N-tile w (16 cols).
// Per-lane base pointers are hoisted so the k-loop is pure
// global_load_b64 (immediate offset) + v_wmma.
// ---------------------------------------------------------------------------
__global__ void node_init_mlp(const float* __restrict__ Uvec,
                              const float* __restrict__ Wi1,
                              const float* __restrict__ bi1,
                              const float* __restrict__ Wi2,
                              const float* __restrict__ bi2,
                              float* __restrict__ x0) {
  __shared__ float partial[16 * 16];   // [wave][sample-in-tile]

  const int lane    = threadIdx.x & 31;
  const int wave    = threadIdx.x >> 5;     // N-tile id, 0..15
  const int tileM   = blockIdx.x * 16;
  const int halfsel = lane >> 4;            // 0: K/M low half, 1: high half
  const int l15     = lane & 15;
  const int koff    = halfsel * 2;
  const int mrow    = tileM + l15;          // A-operand row for this lane
  const int ncol    = wave * 16 + l15;      // B/C/D column for this lane

  // Hoisted 64-bit base pointers (8-byte aligned: all offsets are even).
  const float* ap = Uvec + (size_t)mrow * D_INIT + koff;
  const float* bp = Wi1  + (size_t)ncol * D_INIT + koff;   // B[k][n] = Wi1[n][k]

  v8f c = {0.f, 0.f, 0.f, 0.f, 0.f, 0.f, 0.f, 0.f};
#pragma unroll
  for (int k0 = 0; k0 < D_INIT; k0 += 4) {
    const v2f a  = *(const v2f*)(ap + k0);
    const v2f bb = *(const v2f*)(bp + k0);
    c = __builtin_amdgcn_wmma_f32_16x16x4_f32(
        /*neg_a=*/false, a, /*neg_b=*/false, bb,
        /*c_mod=*/(short)0, c, /*reuse_a=*/false, /*reuse_b=*/false);
  }

  const float bias = bi1[ncol];
  const float w2   = Wi2[ncol];
  float v[8];
#pragma unroll
  for (int r = 0; r < 8; ++r) v[r] = tanhf(c[r] + bias) * w2;

  // Reduce across the 16 N-lanes of each half-wave (xor masks stay in-half).
#pragma unroll
  for (int m = 1; m <= 8; m <<= 1) {
#pragma unroll
    for (int r = 0; r < 8; ++r) v[r] += __shfl_xor(v[r], m, 32);
  }
  if (l15 == 0) {
#pragma unroll
    for (int r = 0; r < 8; ++r) partial[wave * 16 + halfsel * 8 + r] = v[r];
  }
  __syncthreads();

  if (threadIdx.x < 16) {
    float s = 0.f;
#pragma unroll
    for (int w = 0; w < 16; ++w) s += partial[w * 16 + (int)threadIdx.x];
    x0[tileM + threadIdx.x] = s + bi2[0];
  }
}

// ---------------------------------------------------------------------------
// Kernel 3: G[b][n] = sum_k Uvec[b][k] * Wd1[n][2+k] + bd1[n]   (2048 x 512)
// grid = 128, block = 512 (16 waves); wave w owns N-tiles {w, w+16}.
// ---------------------------------------------------------------------------
__global__ void node_g_gemm(const float* __restrict__ Uvec,
                            const float* __restrict__ Wd1,
                            const float* __restrict__ bd1,
                            float* __restrict__ G) {
  const int lane    = threadIdx.x & 31;
  const int wave    = threadIdx.x >> 5;
  const int tileM   = blockIdx.x * 16;
  const int halfsel = lane >> 4;
  const int l15     = lane & 15;
  const int koff    = halfsel * 2;
  const int mrow    = tileM + l15;

  const float* ap = Uvec + (size_t)mrow * D_INIT + koff;   // shared by halves

  for (int half = 0; half < 2; ++half) {
    const int nt   = wave + half * 16;     // N-tile 0..31
    const int ncol = nt * 16 + l15;
    // B[k][n] = Wd1[n][2+k]; base offset 2+koff is even -> 8B aligned.
    const float* bp = Wd1 + (size_t)ncol * D_DYN + 2 + koff;

    v8f c = {0.f, 0.f, 0.f, 0.f, 0.f, 0.f, 0.f, 0.f};
#pragma unroll
    for (int k0 = 0; k0 < D_INIT; k0 += 4) {
      const v2f a  = *(const v2f*)(ap + k0);
      const v2f bb = *(const v2f*)(bp + k0);
      c = __builtin_amdgcn_wmma_f32_16x16x4_f32(
          false, a, false, bb, (short)0, c, false, false);
    }

    const float bias = bd1[ncol];
#pragma unroll
    for (int r = 0; r < 8; ++r) {
      const int m = tileM + halfsel * 8 + r;     // C/D row for vgpr r
      G[(size_t)m * H_DYN + ncol] = c[r] + bias;
    }
  }
}

// ---------------------------------------------------------------------------
// Kernel 4: RK4 time loop, one wave per sample. Sample-independent weight
// columns (wt = Wd1[:,0], wx = Wd1[:,1], w2 = Wd2) are staged into LDS once
// per block with GLOBAL_LOAD_ASYNC_TO_LDS_B32, then each lane keeps its 16
// hidden units (n = lane + 32*i) in registers. Full-wave xor reduce per
// dyn evaluation; 124 serially-dependent stages hidden by 2048 waves.
// grid = B/8 = 256, block = 256 (8 waves = 8 samples).
// ---------------------------------------------------------------------------
__global__ void node_ode(const float* __restrict__ G,
                         const float* __restrict__ x0,
                         const float* __restrict__ b_t_n,
                         const float* __restrict__ Wd1,
                         const float* __restrict__ Wd2,
                         const float* __restrict__ bd2p,
                         float* __restrict__ out) {
  __shared__ float smem[3 * H_DYN];    // wt[512] | wx[512] | w2[512]

  // Async-stage the shared weight columns (6 KB) into LDS.
  for (int idx = threadIdx.x; idx < H_DYN; idx += 256) {
    async_f32_to_lds((uint32_t)(uintptr_t)&smem[idx],
                     Wd1 + (size_t)idx * D_DYN + 0);
    async_f32_to_lds((uint32_t)(uintptr_t)&smem[H_DYN + idx],
                     Wd1 + (size_t)idx * D_DYN + 1);
    async_f32_to_lds((uint32_t)(uintptr_t)&smem[2 * H_DYN + idx],
                     Wd2 + idx);
  }
  wait_asynccnt0();
  __syncthreads();

  const int lane = threadIdx.x & 31;
  const int wave = threadIdx.x >> 5;
  const int b    = blockIdx.x * 8 + wave;

  float g[16], wt[16], wx[16], w2[16];
#pragma unroll
  for (int i = 0; i < 16; ++i) {
    const int n = lane + 32 * i;       // consecutive lanes -> consecutive banks
    g[i]  = G[(size_t)b * H_DYN + n];  // includes bd1; coalesced
    wt[i] = smem[n];
    wx[i] = smem[H_DYN + n];
    w2[i] = smem[2 * H_DYN + n];
  }

  const float sc  = b_t_n[b];
  const float bd2 = bd2p[0];
  float x = x0[b];
  const float h = 1.0f / (float)(STEPS - 1);

  auto dyn = [&](float tt, float xx) -> float {
    const float a = sc * tt;
    float acc = 0.f;
#pragma unroll
    for (int i = 0; i < 16; ++i) {
      const float p = fmaf(xx, wx[i], fmaf(a, wt[i], g[i]));
      acc = fmaf(tanhf(p), w2[i], acc);
    }
#pragma unroll
    for (int m = 1; m < 32; m <<= 1) acc += __shfl_xor(acc, m, 32);
    return (acc + bd2) * sc;
  };

  for (int s = 0; s < STEPS - 1; ++s) {
    const float t  = (float)s * h;
    const float k1 = dyn(t, x);
    const float k2 = dyn(t + 0.5f * h, fmaf(0.5f * h, k1, x));
    const float k3 = dyn(t + 0.5f * h, fmaf(0.5f * h, k2, x));
    const float k4 = dyn(t + h,        fmaf(h, k3, x));
    x = fmaf(h / 6.0f, k1 + 2.0f * k2 + 2.0f * k3 + k4, x);
  }

  if (lane == 0) out[b] = x;
}

// ---------------------------------------------------------------------------
// Host launcher
// inputs: 0 b_i_n(int), 1 b_t_n, 2 U0, 3 U1, 4 U2, 5 Wi1, 6 bi1, 7 Wi2,
//         8 bi2, 9 Wd1, 10 bd1, 11 Wd2, 12 bd2
// ws layout (floats): Uvec[2048*96] | G[2048*512] | x0[2048]  (~4.8 MB)
// ---------------------------------------------------------------------------
extern "C" void kernel_launch(void* const* d_in, const int* in_sizes, int n_in,
                              void* d_out, int out_size, void* d_ws, size_t ws_size,
                              hipStream_t stream) {
  (void)in_sizes; (void)n_in; (void)out_size; (void)ws_size;

  const int*   b_i_n = (const int*)  d_in[0];
  const float* b_t_n = (const float*)d_in[1];
  const float* U0    = (const float*)d_in[2];
  const float* U1    = (const float*)d_in[3];
  const float* U2    = (const float*)d_in[4];
  const float* Wi1   = (const float*)d_in[5];
  const float* bi1   = (const float*)d_in[6];
  const float* Wi2   = (const float*)d_in[7];
  const float* bi2   = (const float*)d_in[8];
  const float* Wd1   = (const float*)d_in[9];
  const float* bd1   = (const float*)d_in[10];
  const float* Wd2   = (const float*)d_in[11];
  const float* bd2   = (const float*)d_in[12];

  float* ws   = (float*)d_ws;
  float* Uvec = ws;                                  // 2048*96
  float* G    = Uvec + (size_t)B_N * D_INIT;         // 2048*512
  float* x0   = G    + (size_t)B_N * H_DYN;          // 2048
  float* out  = (float*)d_out;

  node_gather  <<<B_N,       96, 0, stream>>>(b_i_n, U0, U1, U2, Uvec);
  node_init_mlp<<<B_N / 16, 512, 0, stream>>>(Uvec, Wi1, bi1, Wi2, bi2, x0);
  node_g_gemm  <<<B_N / 16, 512, 0, stream>>>(Uvec, Wd1, bd1, G);
  node_ode     <<<B_N / 8,  256, 0, stream>>>(G, x0, b_t_n, Wd1, Wd2, bd2, out);
}